// SumMarginalHead_30983894073665
// MI455X (gfx1250) — compile-verified
//
#include <hip/hip_runtime.h>
#include <hip/hip_bf16.h>
#include <math.h>

// Problem constants (match reference)
#define BATCH 2048
#define NHEADS 8
#define DDIM 1024
#define VDIM 4096
#define LN_EPS 1e-5f

typedef __attribute__((ext_vector_type(2))) float v2f;
typedef __attribute__((ext_vector_type(8))) float v8f;

// ---------------------------------------------------------------------------
// Kernel A: build h[b,n,d] = gelu(layernorm(input_emb[b] + sum_{k<n} table[k,feat[b,k]]))
// One block per b, 256 threads, 4 elements of D per thread (float4).
// ---------------------------------------------------------------------------
__device__ __forceinline__ float gelu_exact(float x) {
    return 0.5f * x * (1.0f + erff(x * 0.70710678118654752f));
}

__launch_bounds__(256)
__global__ void build_h_kernel(const float* __restrict__ inp,     // (B, D)
                               const int*   __restrict__ feats,   // (B, N)
                               const float* __restrict__ tables,  // (N, V, D)
                               const float* __restrict__ gamma,   // (D,)
                               const float* __restrict__ beta,    // (D,)
                               float*       __restrict__ hout)    // (B, N, D)
{
    const int b  = blockIdx.x;
    const int t  = threadIdx.x;          // 256 threads = 8 wave32
    const int d0 = t * 4;

    __shared__ float red[16];            // [0..7]=sum partials, [8..15]=sumsq partials

    float4 acc = *(const float4*)(inp + (size_t)b * DDIM + d0);
    const float4 g4 = *(const float4*)(gamma + d0);
    const float4 e4 = *(const float4*)(beta  + d0);

    for (int n = 0; n < NHEADS; ++n) {
        // --- block reduction for mean / var over D=1024 ---
        float s  = acc.x + acc.y + acc.z + acc.w;
        float s2 = acc.x*acc.x + acc.y*acc.y + acc.z*acc.z + acc.w*acc.w;
        #pragma unroll
        for (int off = 16; off > 0; off >>= 1) {
            s  += __shfl_xor(s,  off, 32);
            s2 += __shfl_xor(s2, off, 32);
        }
        if ((t & 31) == 0) { red[t >> 5] = s; red[8 + (t >> 5)] = s2; }
        __syncthreads();
        float S = 0.0f, S2 = 0.0f;
        #pragma unroll
        for (int i = 0; i < 8; ++i) { S += red[i]; S2 += red[8 + i]; }
        __syncthreads();   // red[] reused next iteration

        const float mu   = S * (1.0f / DDIM);
        const float var  = S2 * (1.0f / DDIM) - mu * mu;
        const float rstd = rsqrtf(var + LN_EPS);

        float4 hv;
        hv.x = gelu_exact((acc.x - mu) * rstd * g4.x + e4.x);
        hv.y = gelu_exact((acc.y - mu) * rstd * g4.y + e4.y);
        hv.z = gelu_exact((acc.z - mu) * rstd * g4.z + e4.z);
        hv.w = gelu_exact((acc.w - mu) * rstd * g4.w + e4.w);
        *(float4*)(hout + ((size_t)b * NHEADS + n) * DDIM + d0) = hv;

        // --- advance prefix sum with this slot's embedding ---
        const int f = feats[b * NHEADS + n] & (VDIM - 1);
        const float4 e = *(const float4*)(tables + ((size_t)n * VDIM + f) * DDIM + d0);
        acc.x += e.x; acc.y += e.y; acc.z += e.z; acc.w += e.w;
    }
}

// ---------------------------------------------------------------------------
// Kernel B: logits[b,n,v] = sum_d h[b,n,d] * W[n,d,v] + bias[n,v]
// Grid: (V/128, B/128, N). Block: 256 threads = 8 waves.
// Each wave computes a 16(M) x 128(V) tile with V_WMMA_F32_16X16X4_F32.
// W chunk (32 x 128) is staged in LDS with stride 136 (conflict-free halves).
// ---------------------------------------------------------------------------
#define LDS_STRIDE 136   // 2*136 % 64 == 16 -> half-waves hit disjoint bank sets

__launch_bounds__(256)
__global__ void head_gemm_kernel(const float* __restrict__ h,    // (B, N, D)
                                 const float* __restrict__ Wp,   // (N, D, V)
                                 const float* __restrict__ bp,   // (N, V)
                                 float*       __restrict__ out)  // (B, N, V)
{
    const int n     = blockIdx.z;
    const int vbase = blockIdx.x * 128;
    const int mb0   = blockIdx.y * 128;
    const int t     = threadIdx.x;
    const int wave  = t >> 5;
    const int lane  = t & 31;
    const int hi    = lane >> 4;     // half-wave select (K offset +2)
    const int lo    = lane & 15;     // M row / N col within tile

    __shared__ float lds[32 * LDS_STRIDE];

    const int    mbase = mb0 + wave * 16;
    const float* arow  = h  + ((size_t)(mbase + lo) * NHEADS + n) * DDIM;
    const float* wbase = Wp + (size_t)n * DDIM * VDIM + vbase;

    // accumulators initialized with the bias (same column for both half-waves,
    // broadcast over all 8 row-VGPRs)
    v8f acc[8];
    #pragma unroll
    for (int ct = 0; ct < 8; ++ct) {
        const float bias = bp[n * VDIM + vbase + ct * 16 + lo];
        acc[ct] = (v8f){bias, bias, bias, bias, bias, bias, bias, bias};
    }

    for (int kc = 0; kc < DDIM; kc += 32) {
        __syncthreads();
        // --- cooperative stage of W[kc..kc+32) x [vbase..vbase+128) ---
        #pragma unroll
        for (int i = 0; i < 4; ++i) {
            const int s   = t + i * 256;       // float4 slot in 32x128 tile
            const int row = s >> 5;            // 32 float4 per row
            const int c4  = (s & 31) * 4;
            const float4 w = *(const float4*)(wbase + (size_t)(kc + row) * VDIM + c4);
            *(float4*)(&lds[row * LDS_STRIDE + c4]) = w;
        }
        __syncthreads();

        // prefetch next W chunk into L2 while we compute on this one
        if (kc + 32 < DDIM) {
            __builtin_prefetch((const void*)(wbase + (size_t)(kc + 32 + (t >> 3)) * VDIM
                                             + (size_t)(t & 7) * 16), 0, 1);
        }

        // --- A fragments: ISA 16x4 f32 layout: vgpr j, half hi -> K = 4ks + j + 2hi
        v2f af[8];
        #pragma unroll
        for (int ks = 0; ks < 8; ++ks) {
            af[ks] = *(const v2f*)(arow + kc + ks * 4 + 2 * hi);
        }

        #pragma unroll
        for (int ks = 0; ks < 8; ++ks) {
            const float* bptr = &lds[(ks * 4 + 2 * hi) * LDS_STRIDE + lo];
            #pragma unroll
            for (int ct = 0; ct < 8; ++ct) {
                v2f bf;
                bf.x = bptr[ct * 16];                 // K = 4ks + 2hi
                bf.y = bptr[ct * 16 + LDS_STRIDE];    // K = 4ks + 2hi + 1
                acc[ct] = __builtin_amdgcn_wmma_f32_16x16x4_f32(
                    /*neg_a=*/false, af[ks],
                    /*neg_b=*/false, bf,
                    /*c_mod=*/(short)0, acc[ct],
                    /*reuse_a=*/false, /*reuse_b=*/false);
            }
        }
    }

    // --- store: C layout vgpr r -> M = r + 8*hi, col = lo ---
    #pragma unroll
    for (int ct = 0; ct < 8; ++ct) {
        #pragma unroll
        for (int r = 0; r < 8; ++r) {
            const int m = mbase + r + 8 * hi;
            out[((size_t)m * NHEADS + n) * VDIM + vbase + ct * 16 + lo] = acc[ct][r];
        }
    }
}

// ---------------------------------------------------------------------------
extern "C" void kernel_launch(void* const* d_in, const int* in_sizes, int n_in,
                              void* d_out, int out_size, void* d_ws, size_t ws_size,
                              hipStream_t stream) {
    const float* inp    = (const float*)d_in[0];  // (B, D)
    const int*   feats  = (const int*)d_in[1];    // (B, N)
    const float* tables = (const float*)d_in[2];  // (N, V, D)
    const float* Wp     = (const float*)d_in[3];  // (N, D, V)
    const float* bp     = (const float*)d_in[4];  // (N, V)
    const float* gamma  = (const float*)d_in[5];  // (D,)
    const float* beta   = (const float*)d_in[6];  // (D,)
    float*       out    = (float*)d_out;          // (B, N, V)

    float* hbuf = (float*)d_ws;                   // B*N*D floats = 64 MB scratch

    build_h_kernel<<<dim3(BATCH), dim3(256), 0, stream>>>(
        inp, feats, tables, gamma, beta, hbuf);

    head_gemm_kernel<<<dim3(VDIM / 128, BATCH / 128, NHEADS), dim3(256), 0, stream>>>(
        hbuf, Wp, bp, out);
}